// Crf_78924319031484
// MI455X (gfx1250) — compile-verified
//
#include <hip/hip_runtime.h>

typedef __attribute__((ext_vector_type(2))) float v2f;
typedef __attribute__((ext_vector_type(8))) float v8f;

#define B_  64
#define T_  512
#define F_  1024
#define K_  64
#define FC  32                 // f-dim chunk staged in LDS
#define RPB 128                // rows of text per block (8 waves x 16 rows)
#define NCHUNK (F_ / FC)       // 32

// ---------------------------------------------------------------------------
// Kernel 1: emission = text_vec @ W^T + b   via V_WMMA_F32_16X16X4_F32
// Double-buffered GLOBAL_LOAD_ASYNC_TO_LDS_B128 staging (ASYNCcnt-tracked,
// VGPR-bypassing) so HBM streaming overlaps WMMA compute. Memory-bound:
// 142 MB @ 23.3 TB/s ~= 6 us floor; fp32 WMMA keeps exact precision.
// ---------------------------------------------------------------------------
__global__ __launch_bounds__(256) void crf_emission_gemm(
    const float* __restrict__ text,
    const float* __restrict__ W,
    const float* __restrict__ bias,
    float* __restrict__ em)
{
    __shared__ float ldsA[2][RPB * FC];   // 2 x 16 KB: 128 rows x 32 f
    __shared__ float ldsB[2][K_  * FC];   // 2 x  8 KB:  64 tags x 32 f

    const int tid  = threadIdx.x;
    const int lane = tid & 31;
    const int wv   = tid >> 5;                   // wave 0..7
    const int r0   = blockIdx.x * RPB;

    v8f acc[4] = {};                             // 4 n-tiles of 16 cols each

    const int mrow = wv * 16 + (lane & 15);      // local A row
    const int fhal = (lane >> 4) << 1;           // 0 (lanes 0-15) or 2 (16-31)
    const int ncol = lane & 15;

    // issue 6 async b128 loads (4 text + 2 W) for one f-chunk into buffer b
    auto issue_chunk = [&](int fc, int b) {
        #pragma unroll
        for (int i = 0; i < 4; ++i) {
            int q   = tid + i * 256;             // float4 index in 128x32 tile
            int row = q >> 3;
            int fq  = (q & 7) << 2;
            unsigned long long g =
                (unsigned long long)(text + (size_t)(r0 + row) * F_ + fc + fq);
            unsigned l = (unsigned)(unsigned long long)(const void*)
                             (&ldsA[b][row * FC + fq]);     // flat->LDS: low 32b
            asm volatile("global_load_async_to_lds_b128 %0, %1, off"
                         :: "v"(l), "v"(g) : "memory");
        }
        #pragma unroll
        for (int i = 0; i < 2; ++i) {
            int q   = tid + i * 256;             // float4 index in 64x32 tile
            int kk  = q >> 3;
            int fq  = (q & 7) << 2;
            unsigned long long g =
                (unsigned long long)(W + (size_t)kk * F_ + fc + fq);
            unsigned l = (unsigned)(unsigned long long)(const void*)
                             (&ldsB[b][kk * FC + fq]);
            asm volatile("global_load_async_to_lds_b128 %0, %1, off"
                         :: "v"(l), "v"(g) : "memory");
        }
    };

    int buf = 0;
    issue_chunk(0, 0);

    for (int c = 0; c < NCHUNK; ++c) {
        if (c + 1 < NCHUNK) {
            issue_chunk((c + 1) * FC, buf ^ 1);
            // <=12 outstanding, in-order completion: <=6 left => chunk c done
            asm volatile("s_wait_asynccnt 0x6" ::: "memory");
        } else {
            asm volatile("s_wait_asynccnt 0x0" ::: "memory");
        }
        __syncthreads();

        const float* tA = &ldsA[buf][0];
        const float* tB = &ldsB[buf][0];
        #pragma unroll
        for (int fs = 0; fs < FC; fs += 4) {
            v2f a;
            a.x = tA[mrow * FC + fs + fhal];
            a.y = tA[mrow * FC + fs + fhal + 1];
            #pragma unroll
            for (int nt = 0; nt < 4; ++nt) {
                v2f bf;
                bf.x = tB[(nt * 16 + ncol) * FC + fs + fhal];
                bf.y = tB[(nt * 16 + ncol) * FC + fs + fhal + 1];
                acc[nt] = __builtin_amdgcn_wmma_f32_16x16x4_f32(
                    false, a, false, bf, (short)0, acc[nt], false, false);
            }
        }
        __syncthreads();
        buf ^= 1;
    }

    // epilogue: add bias, store C tiles (M = v + (lane>>4)*8, N = lane&15)
    const int mbase = r0 + wv * 16 + ((lane >> 4) << 3);
    #pragma unroll
    for (int nt = 0; nt < 4; ++nt) {
        float bb = bias[nt * 16 + ncol];
        #pragma unroll
        for (int v = 0; v < 8; ++v) {
            em[(size_t)(mbase + v) * K_ + nt * 16 + ncol] = acc[nt][v] + bb;
        }
    }
}

// ---------------------------------------------------------------------------
// Kernel 2: masked Viterbi. One block of 256 threads per sequence:
// thread = (part p = tid>>6 [wave-uniform], tag k = tid&63). Each part
// reduces 16 of the 64 predecessors j; partials combined in ascending part
// order with strict '>' == jnp.argmax first-max. emission[t+1]/mask[t+1]
// are prefetched one step ahead to hide global latency on the serial chain.
// ---------------------------------------------------------------------------
__global__ __launch_bounds__(256) void crf_viterbi(
    const float* __restrict__ em,      // [B, T, K]
    const int*   __restrict__ mask,    // [B, T]
    const float* __restrict__ startT,  // [K]
    const float* __restrict__ endT,    // [K]
    const float* __restrict__ trans,   // [K, K]
    unsigned char* __restrict__ hist,  // [B, T-1, K] backpointers
    int* __restrict__ pred)            // [B, T]
{
    __shared__ float sTrans[K_ * K_];  // 16 KB, [j][k]: conflict-free in k
    __shared__ float sScore[K_];
    __shared__ float sPV[4][K_];
    __shared__ int   sPI[4][K_];
    __shared__ float sRed[K_];

    const int tid = threadIdx.x;
    const int k   = tid & 63;
    const int p   = tid >> 6;          // 0..3, wave-uniform

    const int b = blockIdx.x;
    const float*   emB   = em   + (size_t)b * T_ * K_;
    const int*     maskB = mask + b * T_;
    unsigned char* histB = hist + (size_t)b * (T_ - 1) * K_;

    for (int i = tid; i < K_ * K_; i += 256) sTrans[i] = trans[i];

    float e_nxt = 0.0f;
    int   m_nxt = 0;
    if (p == 0) {
        sScore[k] = startT[k] + emB[k];
        e_nxt = emB[K_ + k];           // t = 1
        m_nxt = maskB[1];
    }
    __syncthreads();

    const int j0 = p * 16;
    for (int t = 1; t < T_; ++t) {
        const float e_cur = e_nxt;
        const int   m_cur = m_nxt;
        if (p == 0 && t + 1 < T_) {    // prefetch next step early
            e_nxt = emB[(t + 1) * K_ + k];
            m_nxt = maskB[t + 1];
        }

        float best = -__builtin_inff();
        int   bj   = j0;
        #pragma unroll
        for (int jj = 0; jj < 16; ++jj) {      // ascending + strict '>'
            int j = j0 + jj;
            float c = sScore[j] + sTrans[j * K_ + k];
            if (c > best) { best = c; bj = j; }
        }
        sPV[p][k] = best;
        sPI[p][k] = bj;
        __syncthreads();

        if (p == 0) {
            float bv = sPV[0][k];
            int   bi = sPI[0][k];
            #pragma unroll
            for (int pp = 1; pp < 4; ++pp) {   // ascending part order
                float v = sPV[pp][k];
                if (v > bv) { bv = v; bi = sPI[pp][k]; }
            }
            if (m_cur) {
                sScore[k] = bv + e_cur;
                histB[(t - 1) * K_ + k] = (unsigned char)bi;
            } else {
                histB[(t - 1) * K_ + k] = (unsigned char)k;   // identity
            }
        }
        __syncthreads();
    }

    if (p == 0) sRed[k] = sScore[k] + endT[k];
    __syncthreads();

    if (tid == 0) {
        float best = sRed[0];
        int   tag  = 0;
        for (int j = 1; j < K_; ++j)
            if (sRed[j] > best) { best = sRed[j]; tag = j; }
        int* predB = pred + b * T_;
        for (int t = T_ - 1; t >= 1; --t) {
            predB[t] = tag;
            tag = histB[(t - 1) * K_ + tag];
        }
        predB[0] = tag;
    }
}

// ---------------------------------------------------------------------------
extern "C" void kernel_launch(void* const* d_in, const int* in_sizes, int n_in,
                              void* d_out, int out_size, void* d_ws, size_t ws_size,
                              hipStream_t stream)
{
    const float* text  = (const float*)d_in[0];   // [B,T,F]
    const int*   mask  = (const int*)  d_in[1];   // [B,T]
    const float* W     = (const float*)d_in[2];   // [K,F]
    const float* bias  = (const float*)d_in[3];   // [K]
    const float* st    = (const float*)d_in[4];   // [K]
    const float* en    = (const float*)d_in[5];   // [K]
    const float* tr    = (const float*)d_in[6];   // [K,K]

    float* emo  = (float*)d_out;                               // [B,1,T,K]
    int*   pred = (int*)((float*)d_out + (size_t)B_ * T_ * K_);// [B,1,T]
    unsigned char* hist = (unsigned char*)d_ws;                // ~2 MB

    crf_emission_gemm<<<(B_ * T_) / RPB, 256, 0, stream>>>(text, W, bias, emo);
    crf_viterbi<<<B_, 256, 0, stream>>>(emo, mask, st, en, tr, hist, pred);
}